// GCN_37211596652590
// MI455X (gfx1250) — compile-verified
//
#include <hip/hip_runtime.h>
#include <hip/hip_bf16.h>
#include <stddef.h>

// ---------------------------------------------------------------------------
// GCN (3x GraphConv, norm='both', ReLU + dropout between layers) for MI455X.
//   N = 100000 nodes, E = 1600000 edges, d = 64 throughout.
// Per layer:   hW = (h * norm_src) @ W      -> fp32 WMMA 16x16x4
//              agg = scatter_add(hW[src] -> dst)   (fp32 atomics, L2 resident)
//              out = agg * norm_dst + b  [ReLU] [dropout]
// ---------------------------------------------------------------------------

typedef __attribute__((ext_vector_type(2))) float v2f;
typedef __attribute__((ext_vector_type(8))) float v8f;

#define D_FEAT 64

// ---------------- utility kernels ----------------

__global__ void fill_zero_kernel(float* __restrict__ p, long long n) {
    long long i = (long long)blockIdx.x * blockDim.x + threadIdx.x;
    if (i < n) p[i] = 0.0f;
}

__global__ void degree_kernel(const int* __restrict__ src, const int* __restrict__ dst,
                              float* __restrict__ deg_out, float* __restrict__ deg_in,
                              int nE) {
    int e = blockIdx.x * blockDim.x + threadIdx.x;
    if (e < nE) {
        atomicAdd(&deg_out[src[e]], 1.0f);
        atomicAdd(&deg_in[dst[e]], 1.0f);
    }
}

// in-place: deg -> (deg > 0) ? rsqrt(deg) : 0   (deg is integer-valued >= 0)
__global__ void norm_kernel(float* __restrict__ d, long long n) {
    long long i = (long long)blockIdx.x * blockDim.x + threadIdx.x;
    if (i < n) {
        float v = d[i];
        d[i] = (v > 0.0f) ? rsqrtf(v) : 0.0f;
    }
}

// ---------------- fused scale + GEMM (WMMA fp32 16x16x4) ----------------
// out[r, :] = (h[r, :] * norm[r]) @ W      W is 64x64 row-major.
// 256 threads = 8 waves / block; each wave owns a 16-row stripe, 4 N-tiles.

__global__ void __launch_bounds__(256)
gemm_scale_wmma_kernel(const float* __restrict__ h,
                       const float* __restrict__ norm,
                       const float* __restrict__ W,
                       float* __restrict__ out, int n) {
    const int wave = threadIdx.x >> 5;
    const int lane = threadIdx.x & 31;
    const int rowBase = (blockIdx.x * 8 + wave) * 16;

    const int r    = lane & 15;        // M for A-frag, N for B/C-frag
    const int koff = (lane >> 4) * 2;  // 0 (lanes 0-15) or 2 (lanes 16-31)

    // Clamp loads so EXEC stays all-ones for the WMMAs (stores are guarded).
    int row = rowBase + r;
    int rowc = (row < n) ? row : (n - 1);
    const float  s    = norm[rowc];
    const float* hrow = h + (size_t)rowc * D_FEAT;

    v8f acc[4];
    acc[0] = v8f{}; acc[1] = v8f{}; acc[2] = v8f{}; acc[3] = v8f{};

#pragma unroll
    for (int kb = 0; kb < D_FEAT; kb += 4) {
        const int k0 = kb + koff;
        v2f a;
        a.x = hrow[k0]     * s;   // A[M][k0]
        a.y = hrow[k0 + 1] * s;   // A[M][k0+1]
        const float* w0 = W + (size_t)k0 * D_FEAT;  // row k0
#pragma unroll
        for (int nt = 0; nt < 4; ++nt) {
            v2f b;
            b.x = w0[nt * 16 + r];           // B[k0  ][nt*16 + N]
            b.y = w0[D_FEAT + nt * 16 + r];  // B[k0+1][nt*16 + N]
            acc[nt] = __builtin_amdgcn_wmma_f32_16x16x4_f32(
                false, a, false, b, (short)0, acc[nt], false, false);
        }
    }

    // C/D layout: VGPR v -> (M = v, N = lane) for lanes 0-15, (M = v+8) for 16-31
    const int mOff = (lane >> 4) * 8;
#pragma unroll
    for (int v = 0; v < 8; ++v) {
        int rr = rowBase + mOff + v;
        if (rr < n) {
            float* orow = out + (size_t)rr * D_FEAT;
#pragma unroll
            for (int nt = 0; nt < 4; ++nt) orow[nt * 16 + r] = acc[nt][v];
        }
    }
}

// ---------------- edge scatter-add ----------------
// One thread per (edge, 4-float chunk): float4 gather from hW[src], 4 atomics
// into agg[dst]. All traffic stays in the 192 MB L2 (node buffers = 25.6 MB).

__global__ void __launch_bounds__(256)
scatter_kernel(const float* __restrict__ hW,
               const int* __restrict__ src, const int* __restrict__ dst,
               float* __restrict__ agg, int nE) {
    long long t = (long long)blockIdx.x * blockDim.x + threadIdx.x;
    int e = (int)(t >> 4);              // 16 chunks of 4 floats per edge
    if (e >= nE) return;
    int c4 = (int)(t & 15);
    int s = src[e];
    int d = dst[e];
    const float4 v = *reinterpret_cast<const float4*>(hW + (size_t)s * D_FEAT + c4 * 4);
    float* a = agg + (size_t)d * D_FEAT + c4 * 4;
    atomicAdd(a + 0, v.x);
    atomicAdd(a + 1, v.y);
    atomicAdd(a + 2, v.z);
    atomicAdd(a + 3, v.w);
}

// ---------------- fused finalize: norm_dst * agg + bias, ReLU, dropout ----
__device__ __forceinline__ unsigned hash_u32(unsigned x) {
    x ^= x >> 16; x *= 0x7feb352du;
    x ^= x >> 15; x *= 0x846ca68bu;
    x ^= x >> 16; return x;
}

__global__ void __launch_bounds__(256)
finalize_kernel(const float* __restrict__ agg, const float* __restrict__ norm_dst,
                const float* __restrict__ bias, float* __restrict__ out,
                int n, int doRelu, int doDrop, unsigned seed) {
    long long i = (long long)blockIdx.x * blockDim.x + threadIdx.x;
    long long total = (long long)n * D_FEAT;
    if (i >= total) return;
    int row = (int)(i >> 6);
    int f   = (int)(i & 63);
    float v = agg[i] * norm_dst[row] + bias[f];
    if (doRelu) v = fmaxf(v, 0.0f);
    if (doDrop) {
        // deterministic keep-prob 0.5 mask, scale kept values by 1/(1-p)=2
        unsigned hv = hash_u32((unsigned)i * 2654435761u + seed);
        v = (hv & 0x80000000u) ? v * 2.0f : 0.0f;
    }
    out[i] = v;
}

// ---------------- launch ----------------

extern "C" void kernel_launch(void* const* d_in, const int* in_sizes, int n_in,
                              void* d_out, int out_size, void* d_ws, size_t ws_size,
                              hipStream_t stream) {
    const float* x  = (const float*)d_in[0];
    const float* W0 = (const float*)d_in[1];
    const float* b0 = (const float*)d_in[2];
    const float* W1 = (const float*)d_in[3];
    const float* b1 = (const float*)d_in[4];
    const float* W2 = (const float*)d_in[5];
    const float* b2 = (const float*)d_in[6];
    const int*  src = (const int*)d_in[7];
    const int*  dst = (const int*)d_in[8];
    float* out = (float*)d_out;

    const int N = in_sizes[0] / D_FEAT;   // 100000
    const int E = in_sizes[7];            // 1600000

    // Workspace carve (floats): 2N norms + 3 * N*64 feature buffers (~78 MB)
    float* ws = (float*)d_ws;
    float* norm_src = ws;                         // N   (first used as deg_out)
    float* norm_dst = ws + N;                     // N   (first used as deg_in)
    float* hbuf     = ws + 2 * (size_t)N;         // N*64 (layer activations)
    float* hW       = hbuf + (size_t)N * D_FEAT;  // N*64
    float* agg      = hW   + (size_t)N * D_FEAT;  // N*64

    const long long feat_total = (long long)N * D_FEAT;
    const int BT = 256;
    const int gNorm = (int)((2LL * N + BT - 1) / BT);
    const int gEdge = (E + BT - 1) / BT;
    const int gFeat = (int)((feat_total + BT - 1) / BT);
    const int gScat = (int)(((long long)E * 16 + BT - 1) / BT);
    const int gGemm = (N + 8 * 16 - 1) / (8 * 16);   // 8 waves * 16 rows per block

    // Degrees -> norms
    fill_zero_kernel<<<gNorm, BT, 0, stream>>>(norm_src, 2LL * N);
    degree_kernel<<<gEdge, BT, 0, stream>>>(src, dst, norm_src, norm_dst, E);
    norm_kernel<<<gNorm, BT, 0, stream>>>(norm_src, 2LL * N);  // both arrays, contiguous

    // Layer 0: x -> hbuf (ReLU + dropout)
    gemm_scale_wmma_kernel<<<gGemm, BT, 0, stream>>>(x, norm_src, W0, hW, N);
    fill_zero_kernel<<<gFeat, BT, 0, stream>>>(agg, feat_total);
    scatter_kernel<<<gScat, BT, 0, stream>>>(hW, src, dst, agg, E);
    finalize_kernel<<<gFeat, BT, 0, stream>>>(agg, norm_dst, b0, hbuf, N, 1, 1, 0x1234567u);

    // Layer 1: hbuf -> hbuf (ReLU + dropout)
    gemm_scale_wmma_kernel<<<gGemm, BT, 0, stream>>>(hbuf, norm_src, W1, hW, N);
    fill_zero_kernel<<<gFeat, BT, 0, stream>>>(agg, feat_total);
    scatter_kernel<<<gScat, BT, 0, stream>>>(hW, src, dst, agg, E);
    finalize_kernel<<<gFeat, BT, 0, stream>>>(agg, norm_dst, b1, hbuf, N, 1, 1, 0x89abcdeu);

    // Layer 2: hbuf -> d_out (no activation, no dropout)
    gemm_scale_wmma_kernel<<<gGemm, BT, 0, stream>>>(hbuf, norm_src, W2, hW, N);
    fill_zero_kernel<<<gFeat, BT, 0, stream>>>(agg, feat_total);
    scatter_kernel<<<gScat, BT, 0, stream>>>(hW, src, dst, agg, E);
    finalize_kernel<<<gFeat, BT, 0, stream>>>(agg, norm_dst, b2, out, N, 0, 0, 0u);
}